// InterfaceBoundaryLoss_80650895884611
// MI455X (gfx1250) — compile-verified
//
#include <hip/hip_runtime.h>

// InterfaceBoundaryLoss for MI455X / gfx1250 (wave32).
// Strategy: fused gather + per-block reduction (kernel 1, deterministic
// partials into d_ws), then a single-block deterministic final reduction
// (kernel 2). Wave-level reduction uses V_WMMA_F32_16X16X4_F32 as a 64-way
// f32 adder tree (A = per-lane partial pair, B = column selector of ones).

typedef __attribute__((ext_vector_type(2))) float v2f;
typedef __attribute__((ext_vector_type(8))) float v8f;

#ifndef IBL_N
#define IBL_N 2048
#endif
#define IBL_B 8
#define IBL_INVD 2047.0f       // 1/DX = N-1
#define IBL_EOUT 80.0f         // E_OUT (E_IN = 1)

#if __has_builtin(__builtin_amdgcn_wmma_f32_16x16x4_f32)
#define IBL_HAVE_WMMA 1
#else
#define IBL_HAVE_WMMA 0
#endif

// Reduce (s1,s2) across a full wave32.
// Returns per-lane: lanes {0,16} -> sum(s1 over 32 lanes),
//                   lanes {1,17} -> sum(s2 over 32 lanes), other lanes 0.
// Requires EXEC == all ones for the calling wave (callers branch at wave
// granularity only).
__device__ __forceinline__ float wave32_reduce_pair(float s1, float s2) {
  const unsigned lane = threadIdx.x & 31u;
#if IBL_HAVE_WMMA
  // A (16x4 f32, 2 VGPRs): VGPR0 = s1 (K=0 lanes 0-15, K=2 lanes 16-31),
  //                        VGPR1 = s2 (K=1 / K=3).
  v2f a;
  a.x = s1;
  a.y = s2;
  // B (4x16 f32): column 0 = 1 on rows K=0,2 (selects s1);
  //               column 1 = 1 on rows K=1,3 (selects s2); rest 0.
  v2f b;
  b.x = ((lane & 15u) == 0u) ? 1.0f : 0.0f;
  b.y = ((lane & 15u) == 1u) ? 1.0f : 0.0f;
  v8f c = {};
  // D[M][0] = s1[M] + s1[M+16], D[M][1] = s2[M] + s2[M+16], D[M][>=2] = 0
  v8f d = __builtin_amdgcn_wmma_f32_16x16x4_f32(
      /*neg_a=*/false, a, /*neg_b=*/false, b,
      /*c_mod=*/(short)0, c, /*reuse_a=*/false, /*reuse_b=*/false);
  // Per-lane sum over the 8 D VGPRs: lane L<16 holds sum over M=0..7 of
  // column L; lane L>=16 holds sum over M=8..15 of column L-16.
  float p = d[0] + d[1] + d[2] + d[3] + d[4] + d[5] + d[6] + d[7];
  // Fold the two M-halves together.
  return p + __shfl_xor(p, 16, 32);
#else
  // Fallback: classic xor-shuffle tree, same output contract.
  float a1 = s1, a2 = s2;
  for (int off = 16; off > 0; off >>= 1) {
    a1 += __shfl_xor(a1, off, 32);
    a2 += __shfl_xor(a2, off, 32);
  }
  const unsigned col = lane & 15u;
  return (col == 0u) ? a1 : ((col == 1u) ? a2 : 0.0f);
#endif
}

__global__ void ibl_partial_kernel(const float* __restrict__ uin,
                                   const float* __restrict__ uout,
                                   const int* __restrict__ xidx,
                                   const int* __restrict__ yidx,
                                   const float* __restrict__ nxarr,
                                   const float* __restrict__ nyarr,
                                   float* __restrict__ partial,
                                   int M, int total) {
  __shared__ float red1[256];
  __shared__ float red2[256];

  const int tid = threadIdx.x;
  const int stride = gridDim.x * blockDim.x;
  float s1 = 0.0f, s2 = 0.0f;

  for (int i = blockIdx.x * blockDim.x + tid; i < total; i += stride) {
    const int b = i / M;
    const int m = i - b * M;

    const int x = xidx[m];
    const int y = yidx[m];
    const float vnx = nxarr[m];
    const float vny = nyarr[m];

    const size_t base = (size_t)b * (size_t)(IBL_N * IBL_N);
    const float* __restrict__ pin = uin + base;
    const float* __restrict__ pout = uout + base;

    const int c = x * IBL_N + y;
    const float cin = pin[c];
    const float cout = pout[c];

    // x-direction: in uses upwind side, out uses the opposite side.
    const bool px = vnx > 0.0f;
    const float sx = px ? 1.0f : -1.0f;
    const int ox = px ? -IBL_N : IBL_N;
    const float ninx = pin[c + ox];
    const float noutx = pout[c - ox];
    const float gx_in = sx * (cin - ninx) * IBL_INVD;
    const float gx_out = -sx * (cout - noutx) * IBL_INVD;

    // y-direction.
    const bool py = vny > 0.0f;
    const float sy = py ? 1.0f : -1.0f;
    const int oy = py ? -1 : 1;
    const float niny = pin[c + oy];
    const float nouty = pout[c - oy];
    const float gy_in = sy * (cin - niny) * IBL_INVD;
    const float gy_out = -sy * (cout - nouty) * IBL_INVD;

    const float nd_in = gx_in * vnx + gy_in * vny;
    const float nd_out = gx_out * vnx + gy_out * vny;

    const float dmatch = cin - cout;
    const float dflux = nd_in - IBL_EOUT * nd_out;  // E_IN = 1
    s1 = fmaf(dmatch, dmatch, s1);
    s2 = fmaf(dflux, dflux, s2);
  }

  red1[tid] = s1;
  red2[tid] = s2;
  __syncthreads();

  if (tid < 32) {  // wave 0 only: EXEC all-ones within the wave
    float a1 = 0.0f, a2 = 0.0f;
#pragma unroll
    for (int j = 0; j < 8; ++j) {
      a1 += red1[tid + j * 32];
      a2 += red2[tid + j * 32];
    }
    const float tot = wave32_reduce_pair(a1, a2);
    if (tid == 0) partial[2 * blockIdx.x] = tot;
    if (tid == 1) partial[2 * blockIdx.x + 1] = tot;
  }
}

__global__ void ibl_final_kernel(const float* __restrict__ partial,
                                 int nblocks, float* __restrict__ out,
                                 float scale) {
  __shared__ float red1[256];
  __shared__ float red2[256];

  const int tid = threadIdx.x;
  float s1 = 0.0f, s2 = 0.0f;
  for (int j = tid; j < nblocks; j += 256) {
    s1 += partial[2 * j];
    s2 += partial[2 * j + 1];
  }
  red1[tid] = s1;
  red2[tid] = s2;
  __syncthreads();

  if (tid < 32) {
    float a1 = 0.0f, a2 = 0.0f;
#pragma unroll
    for (int j = 0; j < 8; ++j) {
      a1 += red1[tid + j * 32];
      a2 += red2[tid + j * 32];
    }
    const float tot = wave32_reduce_pair(a1, a2);
    // lanes {0,16}: S1 ; lanes {1,17}: S2 ; combine neighbor pair.
    const float both = tot + __shfl_xor(tot, 1, 32);
    if (tid == 0) out[0] = both * scale;  // loss = (S1+S2)/(B*M), WEIGHT=1
  }
}

extern "C" void kernel_launch(void* const* d_in, const int* in_sizes, int n_in,
                              void* d_out, int out_size, void* d_ws,
                              size_t ws_size, hipStream_t stream) {
  const float* uin = (const float*)d_in[0];
  const float* uout = (const float*)d_in[1];
  const int* xidx = (const int*)d_in[2];
  const int* yidx = (const int*)d_in[3];
  const float* nxarr = (const float*)d_in[4];
  const float* nyarr = (const float*)d_in[5];

  const int M = in_sizes[2];
  const int total = IBL_B * M;

  const int threads = 256;
  int blocks = (total + threads - 1) / threads;
  if (blocks < 1) blocks = 1;
  if (blocks > 2048) blocks = 2048;  // grid-stride covers the rest

  float* partial = (float*)d_ws;  // 2 floats per block, << ws_size

  ibl_partial_kernel<<<blocks, threads, 0, stream>>>(
      uin, uout, xidx, yidx, nxarr, nyarr, partial, M, total);

  const float scale = (total > 0) ? (1.0f / (float)total) : 0.0f;
  ibl_final_kernel<<<1, 256, 0, stream>>>(partial, blocks, (float*)d_out,
                                          scale);
}